// ModelTorch_83038897701198
// MI455X (gfx1250) — compile-verified
//
#include <hip/hip_runtime.h>

// ---------------------------------------------------------------------------
// Paged-attention decode (GQA) for MI455X / gfx1250, fp32, wave32.
//   B=32, HQ=32, HKV=8, G=4, D=128, PAGE=16, ~256 pages/seq.
// HBM-bound (~1.07 GB KV read): each KV byte is read exactly once, MACs go
// through V_WMMA_F32_16X16X4_F32, softmax on VALU, flash combine via LDS.
// ---------------------------------------------------------------------------

typedef __attribute__((ext_vector_type(2))) float v2f;
typedef __attribute__((ext_vector_type(8))) float v8f;

#define B_    32
#define HQ_   32
#define HKV_  8
#define D_    128
#define PAGE_ 16
#define G_    4       // HQ / HKV
#define NW    8       // waves per block (256 threads)
#define QSTR  132     // padded LDS row stride for Q (bank-conflict-free)

// ds_swizzle group-of-32, SWAPX16: xor=0x10, or=0, and=0x1f -> offset 0x401F
__device__ __forceinline__ float swap16(float x) {
  return __int_as_float(__builtin_amdgcn_ds_swizzle(__float_as_int(x), 0x401F));
}
__device__ __forceinline__ float rdlane(float x, int l) {
  return __int_as_float(__builtin_amdgcn_readlane(__float_as_int(x), l));
}

__global__ __launch_bounds__(NW * 32)
void paged_attn_decode(const float* __restrict__ q,        // [B,HQ,1,D]
                       const float* __restrict__ kv,       // [NPAGES,2,HKV,PAGE,D]
                       const int*   __restrict__ indptr,   // [B+1]
                       const int*   __restrict__ indices,  // [sum pages]
                       const int*   __restrict__ lastlen,  // [B]
                       float* __restrict__ out)            // [B,HQ,1,D]
{
  __shared__ float qs[G_ * QSTR];            // Q^T staged, pre-scaled
  __shared__ float red_acc[NW][G_][D_];      // per-wave unnormalized PV
  __shared__ float red_m[NW][G_];
  __shared__ float red_l[NW][G_];

  const int h    = blockIdx.x;               // kv head
  const int b    = blockIdx.y;               // batch
  const int tid  = threadIdx.x;
  const int wave = tid >> 5;
  const int lane = tid & 31;
  const int half = lane >> 4;                // 0: lanes 0-15, 1: lanes 16-31

  // ---- stage Q (pre-scaled by 1/sqrt(D)) into LDS -------------------------
  const float scale = 0.0883883476483184405f; // 1/sqrt(128)
  for (int i = tid; i < G_ * D_; i += NW * 32) {
    const int g = i >> 7, d = i & (D_ - 1);
    qs[g * QSTR + d] = q[(b * HQ_ + h * G_ + g) * D_ + d] * scale;
  }
  __syncthreads();

  const int pstart = indptr[b];
  const int npages = indptr[b + 1] - pstart;
  const int llen   = lastlen[b];

  // Per-lane addressing for native WMMA f32 16x16x4 operand layouts.
  // A (16x4, M x K): lane l<16 holds A[l][k0..k0+1], lane l+16 holds A[l][k0+2..k0+3]
  const int krow  = lane & 15;    // K-tile row (kv position)
  const int kcol  = half * 2;     // within-4 d offset
  const int vrow  = half * 2;     // V rows s0+{0,1} (lo) / s0+{2,3} (hi)
  const int vcol  = lane & 15;    // d column within 16-wide tile
  const int qhead = lane & 3;     // B columns replicate the 4 GQA heads

  v8f acc[8];                               // out tile: M=head(dup), N=d (8x16)
  #pragma unroll
  for (int t = 0; t < 8; ++t) acc[t] = {};
  float m = -1e30f, lsum = 0.f;

  for (int p = wave; p < npages; p += NW) {           // wave-uniform loop
    const int    pg = indices[pstart + p];
    const float* Kp = kv + ((size_t)pg * 2 * HKV_ + h) * (PAGE_ * D_);
    const float* Vp = Kp + (size_t)HKV_ * PAGE_ * D_;

    // Prefetch next page's K/V (global_prefetch_b8), 256B per lane.
    if (p + NW < npages) {
      const int   pg2 = indices[pstart + p + NW];
      const char* kn  = (const char*)(kv + ((size_t)pg2 * 2 * HKV_ + h) * (PAGE_ * D_));
      __builtin_prefetch(kn + lane * 256,        0, 3);
      __builtin_prefetch(kn + lane * 256 + 128,  0, 3);
      const char* vn = kn + (size_t)HKV_ * PAGE_ * D_ * sizeof(float);
      __builtin_prefetch(vn + lane * 256,        0, 3);
      __builtin_prefetch(vn + lane * 256 + 128,  0, 3);
    }

    // ---- scores = K . q : D[M=pos(16)][N=head col], 32 k-steps over d -----
    v8f s = {};
    const float* kp = Kp + krow * D_ + kcol;
    const float* qp = qs + qhead * QSTR + kcol;
    #pragma unroll
    for (int d0 = 0; d0 < D_; d0 += 4) {
      const v2f a  = *(const v2f*)(kp + d0);   // global, each K elem read once
      const v2f bq = *(const v2f*)(qp + d0);   // LDS broadcast
      s = __builtin_amdgcn_wmma_f32_16x16x4_f32(false, a, false, bq,
                                                (short)0, s, false, false);
    }

    // ---- truncated last page: mask invalid positions (selects, no EXEC) ---
    if (p == npages - 1) {
      #pragma unroll
      for (int v = 0; v < 8; ++v) {
        const int pos = v + 8 * half;
        s[v] = (pos < llen) ? s[v] : -1e30f;
      }
    }

    // ---- online softmax; head column lives in lanes l and l^16 ------------
    float pm = s[0];
    #pragma unroll
    for (int v = 1; v < 8; ++v) pm = fmaxf(pm, s[v]);
    pm = fmaxf(pm, swap16(pm));
    const float mnew = fmaxf(m, pm);
    const float fac  = __expf(m - mnew);
    float w[8], ps = 0.f;
    #pragma unroll
    for (int v = 0; v < 8; ++v) { w[v] = __expf(s[v] - mnew); ps += w[v]; }
    ps   += swap16(ps);
    lsum  = lsum * fac + ps;
    m     = mnew;

    // rescale accumulators: head g's factor is resident in lane g
    #pragma unroll
    for (int g = 0; g < G_; ++g) {
      const float fg = rdlane(fac, g);
      #pragma unroll
      for (int t = 0; t < 8; ++t) acc[t][g] *= fg;
    }

    // ---- rebuild probabilities into PV A-layout (M=head rows, K=4 pos) ----
    float sw[8];
    #pragma unroll
    for (int v = 0; v < 8; ++v) sw[v] = swap16(w[v]);
    const bool lo = (half == 0);
    v2f aw[4];
    aw[0].x = lo ? w[0]  : sw[2];  aw[0].y = lo ? w[1]  : sw[3];   // s0 = 0
    aw[1].x = lo ? w[4]  : sw[6];  aw[1].y = lo ? w[5]  : sw[7];   // s0 = 4
    aw[2].x = lo ? sw[0] : w[2];   aw[2].y = lo ? sw[1] : w[3];    // s0 = 8
    aw[3].x = lo ? sw[4] : w[6];   aw[3].y = lo ? sw[5] : w[7];    // s0 = 12

    // ---- acc += W . V : 8 d-tiles x 4 k-steps, V straight from global -----
    #pragma unroll
    for (int t = 0; t < 8; ++t) {
      #pragma unroll
      for (int kk = 0; kk < 4; ++kk) {
        const int    s0 = 4 * kk;
        const float* vp = Vp + (s0 + vrow) * D_ + 16 * t + vcol;
        v2f bv;
        bv.x = vp[0];            // row s0+vrow   (coalesced 64B per half)
        bv.y = vp[D_];           // row s0+vrow+1
        acc[t] = __builtin_amdgcn_wmma_f32_16x16x4_f32(false, aw[kk], false, bv,
                                                       (short)0, acc[t], false, false);
      }
    }
  }

  // ---- per-wave partials to LDS ------------------------------------------
  if (lane < 16) {
    #pragma unroll
    for (int t = 0; t < 8; ++t)
      #pragma unroll
      for (int g = 0; g < G_; ++g)
        red_acc[wave][g][16 * t + lane] = acc[t][g];
  }
  if (lane < G_) { red_m[wave][lane] = m; red_l[wave][lane] = lsum; }
  __syncthreads();

  // ---- flash-decoding combine across the 8 waves --------------------------
  for (int i = tid; i < G_ * D_; i += NW * 32) {
    const int g = i >> 7, d = i & (D_ - 1);
    float M = -1e30f;
    #pragma unroll
    for (int wv = 0; wv < NW; ++wv) M = fmaxf(M, red_m[wv][g]);
    float L = 0.f, A = 0.f;
    #pragma unroll
    for (int wv = 0; wv < NW; ++wv) {
      const float e = __expf(red_m[wv][g] - M);
      L += e * red_l[wv][g];
      A += e * red_acc[wv][g][d];
    }
    out[(b * HQ_ + h * G_ + g) * D_ + d] = A / L;
  }
}

extern "C" void kernel_launch(void* const* d_in, const int* in_sizes, int n_in,
                              void* d_out, int out_size, void* d_ws, size_t ws_size,
                              hipStream_t stream) {
  (void)in_sizes; (void)n_in; (void)out_size; (void)d_ws; (void)ws_size;
  const float* q       = (const float*)d_in[0];
  const float* kv      = (const float*)d_in[1];
  const int*   indptr  = (const int*)d_in[2];
  const int*   indices = (const int*)d_in[3];
  const int*   lastlen = (const int*)d_in[4];
  float*       out     = (float*)d_out;

  dim3 grid(HKV_, B_);                 // one block per (kv-head, batch)
  paged_attn_decode<<<grid, NW * 32, 0, stream>>>(q, kv, indptr, indices, lastlen, out);
}